// GCN_45595372815203
// MI455X (gfx1250) — compile-verified
//
#include <hip/hip_runtime.h>
#include <math.h>

typedef __attribute__((ext_vector_type(2))) float v2f;
typedef __attribute__((ext_vector_type(8))) float v8f;

#define LD 64   // feature dim (D == H == 64)
#define OC 10   // final classes

// ---------------- degree / normalization ----------------

__global__ void gcn_zero_f32(float* __restrict__ p, long n) {
  long i = (long)blockIdx.x * blockDim.x + threadIdx.x;
  if (i < n) p[i] = 0.0f;
}

__global__ void gcn_deg_accum(const int* __restrict__ idx, float* __restrict__ deg, int E) {
  int i = blockIdx.x * blockDim.x + threadIdx.x;
  if (i < E) atomicAdd(&deg[idx[i]], 1.0f);
}

__global__ void gcn_inv_sqrt(float* __restrict__ d, int n) {
  int i = blockIdx.x * blockDim.x + threadIdx.x;
  if (i < n) {
    float v = d[i];
    d[i] = (v > 0.0f) ? rsqrtf(fmaxf(v, 1.0f)) : 0.0f;
  }
}

// ---------------- dense GEMM: H = (X * norm_src) @ W  [N x 64] ----------------
// 1 block = 16 output rows; 4 waves, each owns one 16x16 column tile.
// K=64 swept as 16 chained V_WMMA_F32_16X16X4_F32 accumulations.
// A frag (16x4 f32): lanes 0-15 hold K=kb+0,kb+1; lanes 16-31 hold K=kb+2,kb+3.
// B frag (4x16 f32): v0 = row kb+koff, v1 = row kb+koff+1 at col = lane%16.
// D (16x16 f32, 8 VGPRs): VGPR v -> row v + 8*(lane/16), col lane%16.
__global__ void __launch_bounds__(128)
gcn_gemm64(const float* __restrict__ X, const float* __restrict__ W,
           const float* __restrict__ norm, float* __restrict__ H, int N) {
  const int lane  = threadIdx.x & 31;
  const int wave  = threadIdx.x >> 5;
  const int rbase = blockIdx.x * 16;
  const int cbase = wave * 16;
  const int m     = lane & 15;
  const int half  = lane >> 4;
  const int koff  = half * 2;

  int row  = rbase + m;
  int rowc = row < N ? row : N - 1;          // clamp loads, keep EXEC all-ones
  const float ns = norm[rowc];
  const float* __restrict__ xrow = X + (size_t)rowc * LD;

  v8f c = {};
#pragma unroll
  for (int kb = 0; kb < LD; kb += 4) {
    v2f a, b;
    a.x = xrow[kb + koff]     * ns;
    a.y = xrow[kb + koff + 1] * ns;
    b.x = W[(kb + koff)     * LD + cbase + m];
    b.y = W[(kb + koff + 1) * LD + cbase + m];
    c = __builtin_amdgcn_wmma_f32_16x16x4_f32(
        /*neg_a=*/false, a, /*neg_b=*/false, b,
        /*c_mod=*/(short)0, c, /*reuse_a=*/false, /*reuse_b=*/false);
  }

#pragma unroll
  for (int v = 0; v < 8; ++v) {
    int r = rbase + v + 8 * half;
    if (r < N) H[(size_t)r * LD + cbase + m] = c[v];
  }
}

// ---------------- SpMM: agg[dst] += h[src]  (gather + scatter-add) ----------------
// 16 threads per edge, float4 per thread -> 64 floats per edge row.
__global__ void gcn_spmm(const float* __restrict__ H, const int* __restrict__ src,
                         const int* __restrict__ dst, float* __restrict__ AGG, int E) {
  long t = (long)blockIdx.x * blockDim.x + threadIdx.x;
  int e = (int)(t >> 4);
  if (e >= E) return;
  int f = ((int)t & 15) * 4;
  int s = src[e], d = dst[e];
  const float4 val = *(const float4*)(H + (size_t)s * LD + f);
  float* out = AGG + (size_t)d * LD + f;
  atomicAdd(out + 0, val.x);
  atomicAdd(out + 1, val.y);
  atomicAdd(out + 2, val.z);
  atomicAdd(out + 3, val.w);
}

// ---------------- epilogue: out = tanh(agg * norm_dst + b) ----------------
__global__ void gcn_finalize(const float* __restrict__ AGG, const float* __restrict__ nd,
                             const float* __restrict__ bias, float* __restrict__ OUT, int N) {
  long i = (long)blockIdx.x * blockDim.x + threadIdx.x;
  if (i >= (long)N * LD) return;
  int r = (int)(i >> 6), f = (int)(i & 63);
  OUT[i] = tanhf(AGG[i] * nd[r] + bias[f]);
}

// ---------------- final FC: out = H @ Wfc + bfc  [N x 10] ----------------
// 1 wave per 16-row tile; B cols 10..15 zero-padded, stores masked to col<10.
__global__ void __launch_bounds__(32)
gcn_gemm_fc(const float* __restrict__ X, const float* __restrict__ W,
            const float* __restrict__ bias, float* __restrict__ OUT, int N) {
  const int lane  = threadIdx.x & 31;
  const int rbase = blockIdx.x * 16;
  const int m     = lane & 15;
  const int half  = lane >> 4;
  const int koff  = half * 2;

  int row  = rbase + m;
  int rowc = row < N ? row : N - 1;
  const float* __restrict__ xrow = X + (size_t)rowc * LD;

  v8f c = {};
#pragma unroll
  for (int kb = 0; kb < LD; kb += 4) {
    v2f a, b;
    a.x = xrow[kb + koff];
    a.y = xrow[kb + koff + 1];
    b.x = (m < OC) ? W[(kb + koff)     * OC + m] : 0.0f;
    b.y = (m < OC) ? W[(kb + koff + 1) * OC + m] : 0.0f;
    c = __builtin_amdgcn_wmma_f32_16x16x4_f32(
        false, a, false, b, (short)0, c, false, false);
  }

  if (m < OC) {
    float bv = bias[m];
#pragma unroll
    for (int v = 0; v < 8; ++v) {
      int r = rbase + v + 8 * half;
      if (r < N) OUT[(size_t)r * OC + m] = c[v] + bv;
    }
  }
}

// ---------------- driver ----------------

extern "C" void kernel_launch(void* const* d_in, const int* in_sizes, int n_in,
                              void* d_out, int out_size, void* d_ws, size_t ws_size,
                              hipStream_t stream) {
  const float* x   = (const float*)d_in[0];
  const int*   src = (const int*)  d_in[1];
  const int*   dst = (const int*)  d_in[2];
  const float* W1  = (const float*)d_in[3];
  const float* b1  = (const float*)d_in[4];
  const float* W2  = (const float*)d_in[5];
  const float* b2  = (const float*)d_in[6];
  const float* W3  = (const float*)d_in[7];
  const float* b3  = (const float*)d_in[8];
  const float* Wfc = (const float*)d_in[9];
  const float* bfc = (const float*)d_in[10];

  const int N = in_sizes[0] / LD;
  const int E = in_sizes[1];
  (void)n_in; (void)out_size; (void)ws_size;

  // workspace carve-out (all offsets 256B aligned)
  char* ws = (char*)d_ws;
  size_t off = 0;
  auto carve = [&](size_t bytes) {
    void* p = ws + off;
    off = (off + bytes + 255) & ~(size_t)255;
    return p;
  };
  float* norm_src = (float*)carve((size_t)N * sizeof(float));
  float* norm_dst = (float*)carve((size_t)N * sizeof(float));
  float* hbuf     = (float*)carve((size_t)N * LD * sizeof(float));
  float* aggbuf   = (float*)carve((size_t)N * LD * sizeof(float));
  float* featbuf  = (float*)carve((size_t)N * LD * sizeof(float));

  // degree normalizations (norm='both')
  gcn_zero_f32 <<<(N + 255) / 256, 256, 0, stream>>>(norm_src, N);
  gcn_zero_f32 <<<(N + 255) / 256, 256, 0, stream>>>(norm_dst, N);
  gcn_deg_accum<<<(E + 255) / 256, 256, 0, stream>>>(src, norm_src, E);
  gcn_deg_accum<<<(E + 255) / 256, 256, 0, stream>>>(dst, norm_dst, E);
  gcn_inv_sqrt <<<(N + 255) / 256, 256, 0, stream>>>(norm_src, N);
  gcn_inv_sqrt <<<(N + 255) / 256, 256, 0, stream>>>(norm_dst, N);

  const int  rtiles = (N + 15) / 16;
  const long NF     = (long)N * LD;
  const long spmm_t = (long)E * 16;

  const float* Ws[3] = {W1, W2, W3};
  const float* bs[3] = {b1, b2, b3};
  const float* in = x;
  for (int l = 0; l < 3; ++l) {
    gcn_gemm64  <<<rtiles, 128, 0, stream>>>(in, Ws[l], norm_src, hbuf, N);
    gcn_zero_f32<<<(int)((NF + 255) / 256), 256, 0, stream>>>(aggbuf, NF);
    gcn_spmm    <<<(int)((spmm_t + 255) / 256), 256, 0, stream>>>(hbuf, src, dst, aggbuf, E);
    gcn_finalize<<<(int)((NF + 255) / 256), 256, 0, stream>>>(aggbuf, norm_dst, bs[l], featbuf, N);
    in = featbuf;
  }
  gcn_gemm_fc<<<rtiles, 32, 0, stream>>>(featbuf, Wfc, bfc, (float*)d_out, N);
}